// MultiCounter_13022340842143
// MI455X (gfx1250) — compile-verified
//
#include <hip/hip_runtime.h>
#include <hip/hip_bf16.h>
#include <cstdint>

// ---------------------------------------------------------------------------
// MultiCounter cumulative-sum:
//   out[t, c] = sum_{s<=t} delta[c, seq[s]]      (L=1M steps, C=64 counters)
//
// 256 MB f32 output dominates (HBM floor ~11us @ 23.3 TB/s). delta (12.9MB)
// + seq (4MB) are L2-resident on MI455X (192MB L2). Plan:
//   0) LDS-tiled transpose delta[64][50257] -> deltaT[50257][64]  (scratch)
//   1) per-chunk column partial sums (1024 chunks of 1024 steps)
//   2) exclusive scan of the 1024 partial vectors (1 tiny block)
//   3) per-chunk scan: 16-step tiles done as Ltri(16x16) x X(16x16) via
//      chained V_WMMA_F32_16X16X4_F32 (exact f32), carry folded into C init.
// ---------------------------------------------------------------------------

typedef __attribute__((ext_vector_type(2))) float v2f;
typedef __attribute__((ext_vector_type(8))) float v8f;

#define MC_L        1048576
#define MC_NI       50257
#define MC_C        64
#define MC_CHUNK    1024
#define MC_NCHUNK   (MC_L / MC_CHUNK)   // 1024

// ---- Pass 0: delta[64][NI] -> deltaT[NI][64], LDS 64x64 tiles -------------
__global__ __launch_bounds__(256) void mc_transpose(const float* __restrict__ delta,
                                                    float* __restrict__ dT) {
    __shared__ float tile[64][65];
    const int i0 = blockIdx.x * 64;
    const int t  = threadIdx.x;
#pragma unroll
    for (int l = 0; l < 16; ++l) {
        int e = l * 256 + t;
        int c = e >> 6;          // 0..63
        int i = e & 63;          // 0..63 (contiguous per half-warp -> coalesced read)
        float v = 0.0f;
        if (i0 + i < MC_NI) v = delta[(size_t)c * MC_NI + (i0 + i)];
        tile[i][c] = v;
    }
    __syncthreads();
#pragma unroll
    for (int l = 0; l < 16; ++l) {
        int e = l * 256 + t;
        int i = e >> 6;
        int c = e & 63;          // contiguous per warp -> coalesced write
        if (i0 + i < MC_NI) dT[(size_t)(i0 + i) * 64 + c] = tile[i][c];
    }
}

// ---- Pass 1: per-chunk column sums ----------------------------------------
__global__ __launch_bounds__(64) void mc_partial(const int* __restrict__ seq,
                                                 const float* __restrict__ dT,
                                                 float* __restrict__ pin,
                                                 int sI, int sC) {
    const int b = blockIdx.x;
    const int c = threadIdx.x;                 // channel 0..63
    const int* __restrict__ s = seq + b * MC_CHUNK;
    float acc = 0.0f;
#pragma unroll 4
    for (int t = 0; t < MC_CHUNK; ++t) {
        acc += dT[(size_t)s[t] * sI + (size_t)c * sC];
    }
    pin[b * 64 + c] = acc;
}

// ---- Pass 2: exclusive scan over chunk partials (single tiny block) -------
__global__ __launch_bounds__(64) void mc_scan_partials(const float* __restrict__ pin,
                                                       float* __restrict__ pout,
                                                       int nchunk) {
    const int c = threadIdx.x;
    float run = 0.0f;
    for (int b = 0; b < nchunk; ++b) {
        pout[b * 64 + c] = run;
        run += pin[b * 64 + c];
    }
}

// ---- Pass 3: WMMA triangular-matrix scan within each chunk ----------------
// Block = 128 threads = 4 waves; wave w handles channels [16w, 16w+16).
// Per 16-step tile:  C = carry + Ltri(16x16) * X(16x16)  via 4 chained
// v_wmma_f32_16x16x4_f32 (A = Ltri column slab, B = gathered delta rows).
__global__ __launch_bounds__(128) void mc_scan(const int* __restrict__ seq,
                                               const float* __restrict__ dT,
                                               const float* __restrict__ pexcl,
                                               float* __restrict__ out,
                                               int sI, int sC) {
    const int b    = blockIdx.x;
    const int lane = threadIdx.x & 31;
    const int w    = threadIdx.x >> 5;       // wave 0..3
    const int col  = lane & 15;              // N index (and M index for A)
    const int hi   = (lane >> 4) & 1;        // lane half selects K/M sub-block
    const int ch   = w * 16 + col;           // absolute channel
    const int chunkStart = b * MC_CHUNK;

    float carry = pexcl[b * 64 + ch];        // same value in lanes l and l+16

    for (int tile = 0; tile < MC_CHUNK / 16; ++tile) {
        const int t0 = chunkStart + tile * 16;
        // 16 seq values for this tile (both lane halves load the same 16)
        const int sv = seq[t0 + col];

        v8f cacc;
#pragma unroll
        for (int r = 0; r < 8; ++r) cacc[r] = carry;   // fold running carry into C

#pragma unroll
        for (int k = 0; k < 4; ++k) {
            // A operand layout (16x4 f32): lanes 0-15 hold K={0,1}, lanes 16-31 K={2,3}
            const int r0 = 4 * k + 2 * hi;   // global K index for VGPR0
            const int r1 = r0 + 1;           // global K index for VGPR1
            v2f a, bm;
            // Ltri[M][K] = (K <= M), M = lane & 15  -> inclusive prefix
            a.x = (r0 <= col) ? 1.0f : 0.0f;
            a.y = (r1 <= col) ? 1.0f : 0.0f;
            // B operand (4x16 f32): same half/K striping; row K of B = X[step r][ch]
            const int g0 = __shfl(sv, r0, 32);
            const int g1 = __shfl(sv, r1, 32);
            bm.x = dT[(size_t)g0 * sI + (size_t)ch * sC];
            bm.y = dT[(size_t)g1 * sI + (size_t)ch * sC];
            cacc = __builtin_amdgcn_wmma_f32_16x16x4_f32(
                false, a, false, bm, (short)0, cacc, false, false);
        }

        // C layout: VGPR r -> (M = r + 8*hi, N = col). Store rows.
#pragma unroll
        for (int r = 0; r < 8; ++r) {
            const int step = t0 + r + 8 * hi;
            out[(size_t)step * 64 + ch] = cacc[r];
        }
        // carry' = C[15][col] (includes old carry); lives in lane 16+col, VGPR7
        float top = cacc[7];
        carry = __shfl(top, 16 + col, 32);
    }
}

extern "C" void kernel_launch(void* const* d_in, const int* in_sizes, int n_in,
                              void* d_out, int out_size, void* d_ws, size_t ws_size,
                              hipStream_t stream) {
    const int*   seq   = (const int*)d_in[0];
    const float* delta = (const float*)d_in[1];
    float*       out   = (float*)d_out;

    const size_t bytesT = (size_t)MC_NI * 64 * sizeof(float);        // ~12.9 MB
    const size_t bytesP = (size_t)MC_NCHUNK * 64 * sizeof(float);    // 256 KB each

    char* ws = (char*)d_ws;
    const float* dT;
    int sI, sC;
    float *pin, *pout;

    if (ws_size >= bytesT + 2 * bytesP) {
        // preferred: transposed delta in scratch -> contiguous 256B gathers
        float* dTe = (float*)ws;
        pin  = (float*)(ws + bytesT);
        pout = pin + MC_NCHUNK * 64;
        mc_transpose<<<(MC_NI + 63) / 64, 256, 0, stream>>>(delta, dTe);
        dT = dTe; sI = 64; sC = 1;
    } else {
        // fallback: gather directly from row-major delta (L2 absorbs strides)
        pin  = (float*)ws;
        pout = pin + MC_NCHUNK * 64;
        dT = delta; sI = 1; sC = MC_NI;
    }

    mc_partial      <<<MC_NCHUNK, 64, 0, stream>>>(seq, dT, pin, sI, sC);
    mc_scan_partials<<<1,          64, 0, stream>>>(pin, pout, MC_NCHUNK);
    mc_scan         <<<MC_NCHUNK, 128, 0, stream>>>(seq, dT, pout, out, sI, sC);
}